// Quanvolution_85667417686518
// MI455X (gfx1250) — compile-verified
//
#include <hip/hip_runtime.h>

// Quanvolution on MI455X (gfx1250):
//   probs = (U_ansatz(params) . (kron_w [RY(2x_w) H]^2 |0>))^2
// U is a fixed 16x16 f32 matrix -> precomputed once (kernel 1); the
// per-patch 16x16 matvec is 4 chained V_WMMA_F32_16X16X4_F32 over tiles of
// 16 patches per wave32 (kernel 2). Workload is store-bandwidth bound
// (~102 MB out @ 23.3 TB/s ~ 4.7 us floor) with instruction issue the
// secondary limit -> division-free geometric grid, hardware v_sin/v_cos,
// folded feature-map math, non-temporal streaming stores.

typedef __attribute__((ext_vector_type(2))) float v2f;
typedef __attribute__((ext_vector_type(8))) float v8f;

#define NQ       4
#define QDIM     16
#define L_LAYERS 3      // AN_REPS + 1
#define AN_REPS  2

// ---------------------------------------------------------------------------
// Kernel 1: build the 16x16 ansatz matrix U, row-major: U[i*16+j] = <i|U|j>.
// Thread j evolves basis state e_j through the ansatz in registers.
// Wire w owns bit (8 >> w)  (wire 0 = MSB, matching the reference reshape).
// Runs once; uses precise libm sin/cos since U feeds every output element.
// ---------------------------------------------------------------------------
__global__ void build_ansatz_U(const float* __restrict__ params,
                               float* __restrict__ U)
{
    const int j = threadIdx.x;
    if (j >= QDIM) return;

    float st[QDIM];
#pragma unroll
    for (int i = 0; i < QDIM; ++i) st[i] = (i == j) ? 1.0f : 0.0f;

    for (int rep = 0; rep < AN_REPS; ++rep) {
        for (int l = 0; l < L_LAYERS; ++l) {
            // RY(params[l][w]) on every wire
#pragma unroll
            for (int w = 0; w < NQ; ++w) {
                const float th = params[l * NQ + w] * 0.5f;
                const float c = cosf(th), s = sinf(th);
                const int bit = 8 >> w;
#pragma unroll
                for (int i = 0; i < QDIM; ++i) {
                    if (i & bit) continue;
                    const float a = st[i], b = st[i + bit];
                    st[i]       = c * a - s * b;
                    st[i + bit] = s * a + c * b;
                }
            }
            // CNOT chain w -> w+1 (pure permutation)
            if (l < L_LAYERS - 1) {
#pragma unroll
                for (int w = 0; w < NQ - 1; ++w) {
                    const int cb = 8 >> w, tb = 8 >> (w + 1);
#pragma unroll
                    for (int i = 0; i < QDIM; ++i) {
                        if ((i & cb) && !(i & tb)) {
                            const float t = st[i];
                            st[i]      = st[i | tb];
                            st[i | tb] = t;
                        }
                    }
                }
            }
        }
    }
#pragma unroll
    for (int i = 0; i < QDIM; ++i) U[i * QDIM + j] = st[i];
}

// ---------------------------------------------------------------------------
// Kernel 2: grid = (Ho rows, Batch); block = 448 threads = 14 waves, wave w
// owns column tile pj in [16w, 16w+16) of its row -> no integer divisions.
//   lane layout (matches documented 16x4 f32 A-matrix striping):
//     n     = lane & 15   -> patch (GEMM N) / A-matrix row (GEMM M)
//     khalf = lane >> 4   -> which K pair of the chunk this lane holds
//   GEMM: D(16x16) = U(16x16) x V(16 x 16patches), K split into 4 WMMA
//   16x16x4 f32 steps. D layout: VGPR r -> out dim (r + 8*khalf), lane -> n.
// Edge (pj == 223): loads clamped so EXEC stays all-1s through the WMMAs;
// only the final stores are predicated.
// ---------------------------------------------------------------------------
__global__ void __launch_bounds__(448)
quanv_wmma(const float* __restrict__ img,
           const float* __restrict__ U,
           float* __restrict__ out)
{
    const int W = 224, Wo = 223;
    const int HoWo = 223 * 223;

    const int lane  = threadIdx.x & 31;
    const int wave  = threadIdx.x >> 5;    // 0..13 : column tile in this row
    const int n     = lane & 15;
    const int khalf = lane >> 4;

    const int pi = blockIdx.x;             // output row   0..222
    const int b  = blockIdx.y;             // batch        0..31

    const int pj    = wave * 16 + n;       // output col (may be 223 once)
    const bool ok   = pj < Wo;
    const int pjc   = ok ? pj : (Wo - 1);  // clamped for loads

    // ---- patch pixels ------------------------------------------------------
    const float* base = img + (b * 224 + pi) * W + pjc;
    const float xs[4] = { base[0], base[1], base[W], base[W + 1] };

    // ---- per-wire 2-vector:  t_w = [RY(2 x_w) . H]^2 |0>  ------------------
    // First H|0> and RY fold to (R(c-s), R(c+s)); x in [0,1) rad -> hardware
    // v_sin/v_cos via __sincosf is accurate here.
    const float RHALF = 0.70710678118654752440f;
    float t[NQ][2];
#pragma unroll
    for (int w = 0; w < NQ; ++w) {
        float s, c;
        __sincosf(xs[w], &s, &c);          // half-angle of RY(2x) is x
        const float a1 = (c - s) * RHALF;  // after H, RY
        const float b1 = (c + s) * RHALF;
        const float ha = (a1 + b1) * RHALF; // second H
        const float hb = (a1 - b1) * RHALF;
        t[w][0] = c * ha - s * hb;          // second RY
        t[w][1] = s * ha + c * hb;
    }

    // ---- 4 chained V_WMMA_F32_16X16X4_F32 over K = 0..15 -------------------
    // k = 4c + 2*khalf + v : wire0 bit = c>>1, wire1 = c&1, wire2 = khalf,
    //                        wire3 = v
    const float u0 = t[2][khalf] * t[3][0];
    const float u1 = t[2][khalf] * t[3][1];
    v8f acc = {};
#pragma unroll
    for (int c = 0; c < 4; ++c) {
        const int kbase = 4 * c + 2 * khalf;
        const v2f Afrag = { U[n * QDIM + kbase], U[n * QDIM + kbase + 1] };
        const float pre = t[0][c >> 1] * t[1][c & 1];
        const v2f Bfrag = { pre * u0, pre * u1 };
        acc = __builtin_amdgcn_wmma_f32_16x16x4_f32(
                  /*neg_a=*/false, Afrag,
                  /*neg_b=*/false, Bfrag,
                  /*c_mod=*/(short)0, acc,
                  /*reuse_a=*/false, /*reuse_b=*/false);
    }

    // ---- square amplitudes, stream to (B, 16, Ho, Wo) non-temporally -------
    if (ok) {
        const int obase = (b * QDIM) * HoWo + pi * Wo + pj;
#pragma unroll
        for (int r = 0; r < 8; ++r) {
            const int d = r + 8 * khalf;
            const float v = acc[r];
            __builtin_nontemporal_store(v * v, out + obase + d * HoWo);
        }
    }
}

// ---------------------------------------------------------------------------
extern "C" void kernel_launch(void* const* d_in, const int* in_sizes, int n_in,
                              void* d_out, int out_size, void* d_ws, size_t ws_size,
                              hipStream_t stream)
{
    (void)in_sizes; (void)n_in; (void)out_size; (void)ws_size;

    const float* img    = (const float*)d_in[0];   // (32,1,224,224) f32
    const float* params = (const float*)d_in[1];   // (3,4) f32
    float*       out    = (float*)d_out;           // (32,16,223,223) f32
    float*       U      = (float*)d_ws;            // 16x16 f32 scratch (1 KB)

    // Stage 1: tiny kernel builds the fixed ansatz matrix.
    build_ansatz_U<<<1, 16, 0, stream>>>(params, U);

    // Stage 2: WMMA quanvolution. One block per (row, batch); 14 waves cover
    // the 223 output columns in 16-wide tiles (division-free indexing).
    dim3 grid(223, 32, 1);
    quanv_wmma<<<grid, 448, 0, stream>>>(img, U, out);
}